// aed_pos_33363305955887
// MI455X (gfx1250) — compile-verified
//
#include <hip/hip_runtime.h>
#include <hip/hip_bf16.h>

typedef float v2f __attribute__((ext_vector_type(2)));
typedef float v8f __attribute__((ext_vector_type(8)));

#define NPIX 9216   // 96*96
#define NB   2      // batch
#define NC   4      // channels
#define TPB  256    // 8 wave32 per block

// ---------------------------------------------------------------------------
// Matrix-core block reduction (wave32, gfx1250).
// Reduces 256 floats (one per thread) using chained V_WMMA_F32_16X16X4_F32
// with an all-ones B matrix: D[m,n] = sum_k A[m,k] + C[m,n]  (row sums,
// broadcast across columns). 4 chained WMMAs consume 4*64 = 256 values.
// D layout (32-bit C/D, 16x16): lane 0 holds M=0..7 col0 in VGPR0..7,
// lane 16 holds M=8..15 col0 -> total = shfl(s,0) + shfl(s,16).
// Only wave 0 issues the WMMA (EXEC all-ones within the wave, as required).
// Result valid for wave 0; thread 0 uses it.
// ---------------------------------------------------------------------------
__device__ __forceinline__ float wmma_block_sum(float v, float* lds) {
    const int tid = threadIdx.x;
    lds[tid] = v;
    __syncthreads();
    float total = 0.0f;
    if (tid < 32) {                         // exactly wave 0, EXEC = all ones
        v8f acc = {};                       // C starts at zero
        v2f bones; bones.x = 1.0f; bones.y = 1.0f;   // all-ones B (4x16)
        #pragma unroll
        for (int t = 0; t < 4; ++t) {
            v2f a;                          // A 16x4 f32: 2 VGPRs per lane
            a.x = lds[t * 64 + tid * 2 + 0];
            a.y = lds[t * 64 + tid * 2 + 1];
            // 8 args: (neg_a, A, neg_b, B, c_mod, C, reuse_a, reuse_b)
            acc = __builtin_amdgcn_wmma_f32_16x16x4_f32(
                false, a, false, bones, (short)0, acc, false, false);
        }
        float s = acc[0] + acc[1] + acc[2] + acc[3] +
                  acc[4] + acc[5] + acc[6] + acc[7];
        total = __shfl(s, 0, 32) + __shfl(s, 16, 32);
    }
    __syncthreads();                        // LDS reusable by caller afterwards
    return total;
}

// ---------------------------------------------------------------------------
// Kernel 1: fused per-pixel math. 18432 pixels, 72 blocks x 256 threads.
// Emits a[] and per-block partials {sumMask, sumL1, sumPull, sumNz}.
// ---------------------------------------------------------------------------
__global__ void elem_kernel(const float* __restrict__ pred,
                            const float* __restrict__ label,
                            float* __restrict__ a_out,
                            float* __restrict__ partials) {
    __shared__ float red[TPB];
    const int p   = blockIdx.x * TPB + threadIdx.x;     // < NB*NPIX exactly
    const int b   = p / NPIX;
    const int idx = p - b * NPIX;

    const float* pb = pred  + (size_t)b * NC * NPIX + idx;
    const float* lb = label + (size_t)b * 2  * NPIX + idx;

    const float p0 = pb[0 * NPIX], p1 = pb[1 * NPIX];
    const float p2 = pb[2 * NPIX], p3 = pb[3 * NPIX];
    const float target = lb[0];
    const float m      = lb[NPIX];

    const float dens = sqrtf(p0 * p0 + p1 * p1 + p2 * p2 + p3 * p3);

    // masked smooth-L1
    const float d  = fabsf(dens - target);
    const float sl = m * ((d < 1.0f) ? 0.5f * d * d : d - 0.5f);

    // diversity / pull
    const float inv = m / (dens + 1e-4f);
    const float d0 = p0 * inv, d1 = p1 * inv, d2 = p2 * inv, d3 = p3 * inv;
    const float aver = (d0 + d1 + d2 + d3) * 0.25f * m;
    const float e0 = d0 - aver, e1 = d1 - aver, e2 = d2 - aver, e3 = d3 - aver;
    const float pullc = e0 * e0 + e1 * e1 + e2 * e2 + e3 * e3;

    const float aval = aver * m;                       // a = aver * mask
    const float nzv  = (aval != 0.0f) ? 1.0f : 0.0f;
    a_out[p] = aval;

    const float t0 = wmma_block_sum(m,     red);
    const float t1 = wmma_block_sum(sl,    red);
    const float t2 = wmma_block_sum(pullc, red);
    const float t3 = wmma_block_sum(nzv,   red);
    if (threadIdx.x == 0) {
        partials[blockIdx.x * 4 + 0] = t0;
        partials[blockIdx.x * 4 + 1] = t1;
        partials[blockIdx.x * 4 + 2] = t2;
        partials[blockIdx.x * 4 + 3] = t3;
    }
}

// ---------------------------------------------------------------------------
// Kernel 2: pairwise push term, LDS-tiled, deterministic (no atomics).
// Block = (batch, 256-row tile); j sweeps all 9216 cols in 256-wide chunks.
// pairSum includes the diagonal (contributes exactly sumNz); final kernel
// subtracts sumNz, matching the reference's  sum(pair_mask*diff) - sum(nz).
// ---------------------------------------------------------------------------
__global__ void pair_kernel(const float* __restrict__ a,
                            float* __restrict__ partials) {
    __shared__ float ajs[TPB];
    __shared__ float red[TPB];
    const int tilesPerBatch = NPIX / TPB;               // 36
    const int b    = blockIdx.x / tilesPerBatch;
    const int tile = blockIdx.x - b * tilesPerBatch;
    const int i    = tile * TPB + threadIdx.x;

    const float* ab = a + (size_t)b * NPIX;
    const float ai  = ab[i];
    const bool  nzi = (ai != 0.0f);

    float acc = 0.0f;
    for (int jb = 0; jb < NPIX; jb += TPB) {
        __syncthreads();
        ajs[threadIdx.x] = ab[jb + threadIdx.x];
        __syncthreads();
        if (nzi) {
            #pragma unroll 16
            for (int jj = 0; jj < TPB; ++jj) {
                const float aj = ajs[jj];
                float t = fmaxf(1.0f - fabsf(ai - aj), 0.0f);
                acc += (aj != 0.0f) ? t : 0.0f;
            }
        }
    }
    __syncthreads();
    const float tot = wmma_block_sum(acc, red);
    if (threadIdx.x == 0) partials[blockIdx.x] = tot;
}

// ---------------------------------------------------------------------------
// Kernel 3: deterministic final combine (single thread; ~360 adds).
// ---------------------------------------------------------------------------
__global__ void final_kernel(const float* __restrict__ elemP,
                             const float* __restrict__ pairP,
                             int nElemBlocks, int nPairBlocks,
                             float* __restrict__ out) {
    if (threadIdx.x == 0 && blockIdx.x == 0) {
        float sm = 0.f, sl = 0.f, sp = 0.f, snz = 0.f, pr = 0.f;
        for (int i = 0; i < nElemBlocks; ++i) {
            sm  += elemP[4 * i + 0];
            sl  += elemP[4 * i + 1];
            sp  += elemP[4 * i + 2];
            snz += elemP[4 * i + 3];
        }
        for (int i = 0; i < nPairBlocks; ++i) pr += pairP[i];
        const float denom    = fmaxf(1.0f, sm);
        const float den_loss = 5.0f * sl / denom;
        const float pull     = sp / denom;
        const float push     = pr - snz;                // diagonal == snz
        out[0] = 0.01f * (den_loss + pull + push);
    }
}

extern "C" void kernel_launch(void* const* d_in, const int* in_sizes, int n_in,
                              void* d_out, int out_size, void* d_ws, size_t ws_size,
                              hipStream_t stream) {
    const float* pred  = (const float*)d_in[0];   // (2,4,96,96) f32
    const float* label = (const float*)d_in[1];   // (2,2,96,96) f32

    float* ws    = (float*)d_ws;
    float* a     = ws;                      // NB*NPIX = 18432 floats
    float* elemP = a + NB * NPIX;           // 72*4 floats
    float* pairP = elemP + 72 * 4;          // 72 floats

    const int nElemBlocks = (NB * NPIX) / TPB;            // 72
    const int nPairBlocks = NB * (NPIX / TPB);            // 72

    elem_kernel <<<nElemBlocks, TPB, 0, stream>>>(pred, label, a, elemP);
    pair_kernel <<<nPairBlocks, TPB, 0, stream>>>(a, pairP);
    final_kernel<<<1, 32, 0, stream>>>(elemP, pairP, nElemBlocks, nPairBlocks,
                                       (float*)d_out);
}